// KDNet_77979426226850
// MI455X (gfx1250) — compile-verified
//
#include <hip/hip_runtime.h>
#include <hip/hip_bf16.h>

typedef float v2f __attribute__((ext_vector_type(2)));
typedef float v8f __attribute__((ext_vector_type(8)));

#define Bsz 512
#define EPSV 1e-5f

// ---------------------------------------------------------------------------
// fc1: h[b,f,n] = sum_c x[b,c,n] * w[c,f] + bias[f]   (3 -> 32 per point)
// ---------------------------------------------------------------------------
__global__ __launch_bounds__(256) void kd_fc1(const float* __restrict__ x,
                                              const float* __restrict__ w,
                                              const float* __restrict__ bias,
                                              float* __restrict__ H) {
  int idx = blockIdx.x * blockDim.x + threadIdx.x;   // 0 .. 32*1024-1
  int b = blockIdx.y;
  int f = idx >> 10;
  int n = idx & 1023;
  const float* xb = x + (size_t)b * 3 * 1024;
  float v0 = xb[n], v1 = xb[1024 + n], v2 = xb[2048 + n];
  H[(size_t)b * 32 * 1024 + idx] =
      fmaf(v0, w[f], fmaf(v1, w[32 + f], fmaf(v2, w[64 + f], bias[f])));
}

// ---------------------------------------------------------------------------
// Per-level GEMM + folded-BN + ReLU via V_WMMA_F32_16X16X4_F32.
// One wave computes a 32x16 tile (two 16x16 M-subtiles sharing one B frag):
//   A 16x4 f32 frag : lane<16 -> K{0,1}, lane>=16 -> K{2,3}; row m = lane&15
//   B 4x16 f32 frag : col n = lane&15; K halves split across lane halves
//   C/D 16x16 frag  : row M = v + 8*(lane>=16), col N = lane&15
// OOB columns (n>=N) are clamped to a safe in-bounds address: a B column
// only feeds its own D column, and those D columns are never stored, so the
// inner loop is completely branch-free (no per-iteration EXEC churn).
// ---------------------------------------------------------------------------
__global__ __launch_bounds__(32) void kd_gemm_bn_relu(
    const float* __restrict__ H,   // [B, K, N]
    const float* __restrict__ W,   // [M, K]
    const float* __restrict__ cb,  // [M] conv bias (zeros, kept for fidelity)
    const float* __restrict__ gg,  // [M] bn gamma
    const float* __restrict__ bb,  // [M] bn beta
    const float* __restrict__ bm,  // [M] bn mean
    const float* __restrict__ bv,  // [M] bn var
    float* __restrict__ Y,         // [B, M, N]  (post BN+ReLU)
    int M, int K, int N, int tilesN) {
  int tile = blockIdx.x;
  int b = blockIdx.y;
  int tm = tile / tilesN;              // covers rows [32*tm, 32*tm+32)
  int tn = tile - tm * tilesN;
  int m0 = tm << 5;
  int n0 = tn << 4;
  int lane = threadIdx.x;
  int half = lane >> 4;                // 0 or 1
  int l = lane & 15;
  int n = n0 + l;
  bool ncol_ok = (n < N);
  int nc = ncol_ok ? n : (N - 1);      // clamp: OOB cols read safe garbage

  const float* Wrow0 = W + (size_t)(m0 + l) * K + half * 2;        // subtile 0
  const float* Wrow1 = Wrow0 + (size_t)16 * K;                     // subtile 1
  const float* Hcol = H + (size_t)b * K * N + (size_t)(half * 2) * N + nc;

  v8f acc0 = {};
  v8f acc1 = {};
  for (int k0 = 0; k0 < K; k0 += 4) {
    v2f bf, a0, a1;
    bf.x = Hcol[(size_t)k0 * N];
    bf.y = Hcol[(size_t)(k0 + 1) * N];
    a0.x = Wrow0[k0];
    a0.y = Wrow0[k0 + 1];
    a1.x = Wrow1[k0];
    a1.y = Wrow1[k0 + 1];
    acc0 = __builtin_amdgcn_wmma_f32_16x16x4_f32(
        false, a0, false, bf, (short)0, acc0, false, false);
    acc1 = __builtin_amdgcn_wmma_f32_16x16x4_f32(
        false, a1, false, bf, (short)0, acc1, false, false);
  }

  if (ncol_ok) {
    float* Yb = Y + (size_t)b * M * N;
#pragma unroll
    for (int v = 0; v < 8; ++v) {
      int r = v + half * 8;
      {
        int mrow = m0 + r;
        float scale = gg[mrow] * rsqrtf(bv[mrow] + EPSV);
        float shift = bb[mrow] - bm[mrow] * scale;
        float val = fmaf(acc0[v] + cb[mrow], scale, shift);
        Yb[(size_t)mrow * N + n] = fmaxf(val, 0.0f);
      }
      {
        int mrow = m0 + 16 + r;
        float scale = gg[mrow] * rsqrtf(bv[mrow] + EPSV);
        float shift = bb[mrow] - bm[mrow] * scale;
        float val = fmaf(acc1[v] + cb[mrow], scale, shift);
        Yb[(size_t)mrow * N + n] = fmaxf(val, 0.0f);
      }
    }
  }
}

// ---------------------------------------------------------------------------
// take_along_axis select + pair-adjacent-points:
//   flat m = 3p + sel[b,p];  c3 = 3c + m/N;  n = m%N
//   out[b, cOut, q] with p = 2q + (cOut>=C), c = cOut%C
// ---------------------------------------------------------------------------
__global__ __launch_bounds__(256) void kd_gather_pair(
    const float* __restrict__ Y,   // [B, 3C, N]
    const int* __restrict__ sel,   // [B, N]
    float* __restrict__ Hout,      // [B, 2C, N/2]
    int C, int N) {
  int halfN = N >> 1;
  int total = 2 * C * halfN;
  int idx = blockIdx.x * blockDim.x + threadIdx.x;
  if (idx >= total) return;
  int b = blockIdx.y;
  int q = idx % halfN;
  int cOut = idx / halfN;
  int p = 2 * q + (cOut >= C ? 1 : 0);
  int c = (cOut >= C) ? (cOut - C) : cOut;
  int m = 3 * p + sel[(size_t)b * N + p];
  int j = m / N;
  int n = m - j * N;
  float vv = Y[((size_t)b * 3 * C + (3 * c + j)) * N + n];
  Hout[((size_t)b * 2 * C + cOut) * halfN + q] = vv;
}

// ---------------------------------------------------------------------------
// Final classifier: scores = h[b,:,0] @ fc8_w + fc8_b ; log_softmax.
// One block per batch, deterministic serial per-class reduction through LDS.
// ---------------------------------------------------------------------------
__global__ __launch_bounds__(128) void kd_classify(
    const float* __restrict__ Hf,   // [B, 2048]
    const float* __restrict__ Wf,   // [2048, 40]
    const float* __restrict__ bf,   // [40]
    float* __restrict__ out) {      // [B, 40]
  __shared__ float part[128][40];
  __shared__ float scores[40];
  __shared__ float red[2];
  int b = blockIdx.x;
  int t = threadIdx.x;
  float acc[40];
#pragma unroll
  for (int k = 0; k < 40; ++k) acc[k] = 0.0f;
  const float* h = Hf + (size_t)b * 2048;
  for (int c = t; c < 2048; c += 128) {
    float hv = h[c];
    const float* wr = Wf + (size_t)c * 40;
#pragma unroll
    for (int k = 0; k < 40; ++k) acc[k] = fmaf(hv, wr[k], acc[k]);
  }
#pragma unroll
  for (int k = 0; k < 40; ++k) part[t][k] = acc[k];
  __syncthreads();
  if (t < 40) {
    float s = bf[t];
    for (int i = 0; i < 128; ++i) s += part[i][t];
    scores[t] = s;
  }
  __syncthreads();
  if (t == 0) {
    float mx = scores[0];
    for (int k = 1; k < 40; ++k) mx = fmaxf(mx, scores[k]);
    float se = 0.0f;
    for (int k = 0; k < 40; ++k) se += expf(scores[k] - mx);
    red[0] = mx;
    red[1] = logf(se);
  }
  __syncthreads();
  if (t < 40) out[(size_t)b * 40 + t] = scores[t] - red[0] - red[1];
}

// ---------------------------------------------------------------------------
extern "C" void kernel_launch(void* const* d_in, const int* in_sizes, int n_in,
                              void* d_out, int out_size, void* d_ws, size_t ws_size,
                              hipStream_t stream) {
  (void)in_sizes; (void)n_in; (void)out_size; (void)ws_size;

  const float* x    = (const float*)d_in[0];
  const int*   sel[10];
  for (int i = 0; i < 10; ++i) sel[i] = (const int*)d_in[1 + i];
  const float* fc1w = (const float*)d_in[11];
  const float* fc1b = (const float*)d_in[12];
  const float* cw[10], *cbi[10], *gG[10], *gB[10], *gM[10], *gV[10];
  for (int i = 0; i < 10; ++i) {
    cw[i]  = (const float*)d_in[13 + i];
    cbi[i] = (const float*)d_in[23 + i];
    gG[i]  = (const float*)d_in[33 + i];
    gB[i]  = (const float*)d_in[43 + i];
    gM[i]  = (const float*)d_in[53 + i];
    gV[i]  = (const float*)d_in[63 + i];
  }
  const float* fc8w = (const float*)d_in[73];
  const float* fc8b = (const float*)d_in[74];

  // workspace: two ping-pong activation buffers + one Y scratch
  const size_t HBUF = (size_t)Bsz * 32 * 1024;   // 16,777,216 floats (max level act)
  float* ws = (float*)d_ws;
  float* Hbuf[2] = { ws, ws + HBUF };
  float* Ytmp = ws + 2 * HBUF;                   // up to 50,331,648 floats

  // fc1: [B,3,1024] -> [B,32,1024]
  {
    dim3 grid((32 * 1024) / 256, Bsz);
    kd_fc1<<<grid, 256, 0, stream>>>(x, fc1w, fc1b, Hbuf[0]);
  }

  static const int CSv[10]   = {32, 64, 64, 128, 128, 256, 256, 512, 512, 1024};
  static const int CINSv[10] = {32, 64, 128, 128, 256, 256, 512, 512, 1024, 1024};
  static const int NSv[10]   = {1024, 512, 256, 128, 64, 32, 16, 8, 4, 2};

  int cur = 0;
  for (int i = 0; i < 10; ++i) {
    int C = CSv[i], K = CINSv[i], N = NSv[i];
    int M = 3 * C;
    int tilesM2 = M / 32;              // 32-row tiles per wave (M/16 is even)
    int tilesN = (N + 15) / 16;
    dim3 g1(tilesM2 * tilesN, Bsz);
    kd_gemm_bn_relu<<<g1, 32, 0, stream>>>(Hbuf[cur], cw[i], cbi[i], gG[i],
                                           gB[i], gM[i], gV[i], Ytmp,
                                           M, K, N, tilesN);
    int halfN = N / 2;
    int total = 2 * C * halfN;
    dim3 g2((total + 255) / 256, Bsz);
    kd_gather_pair<<<g2, 256, 0, stream>>>(Ytmp, sel[i], Hbuf[1 - cur], C, N);
    cur = 1 - cur;
  }

  // classifier + log_softmax -> d_out [B, 40]
  kd_classify<<<Bsz, 128, 0, stream>>>(Hbuf[cur], fc8w, fc8b, (float*)d_out);
}